// DualConv_36240934043743
// MI455X (gfx1250) — compile-verified
//
#include <hip/hip_runtime.h>
#include <cstdint>

typedef float v2f __attribute__((ext_vector_type(2)));
typedef float v8f __attribute__((ext_vector_type(8)));

#define IN_CH   64
#define HEADS   4
#define OUT_CH  32
#define WCOLS   128   // HEADS*OUT_CH

static __device__ __forceinline__ void softmax4(float t0, float t1, float t2, float t3,
                                                float& q0, float& q1, float& q2, float& q3) {
    float m = fmaxf(fmaxf(t0, t1), fmaxf(t2, t3));
    q0 = __expf(t0 - m);
    q1 = __expf(t1 - m);
    q2 = __expf(t2 - m);
    q3 = __expf(t3 - m);
    float inv = 1.0f / (q0 + q1 + q2 + q3);
    q0 *= inv; q1 *= inv; q2 *= inv; q3 *= inv;
}

// ---------------------------------------------------------------------------
// P = X @ U  (N x 4), one thread per node. U (64x4) is broadcast/cached.
// ---------------------------------------------------------------------------
__global__ __launch_bounds__(256) void feast_p_kernel(const float* __restrict__ X,
                                                      const float* __restrict__ U,
                                                      float4* __restrict__ P, int N) {
    int i = blockIdx.x * blockDim.x + threadIdx.x;
    if (i >= N) return;
    const float* xr = X + (size_t)i * IN_CH;
    float a0 = 0.f, a1 = 0.f, a2 = 0.f, a3 = 0.f;
#pragma unroll
    for (int k = 0; k < IN_CH; ++k) {
        float xv = xr[k];
        a0 = fmaf(xv, U[k * 4 + 0], a0);
        a1 = fmaf(xv, U[k * 4 + 1], a1);
        a2 = fmaf(xv, U[k * 4 + 2], a2);
        a3 = fmaf(xv, U[k * 4 + 3], a3);
    }
    P[i] = make_float4(a0, a1, a2, a3);
}

// ---------------------------------------------------------------------------
// Y = X @ W  (N x 128) with V_WMMA_F32_16X16X4_F32 (full fp32 fidelity).
// Block = 256 threads = 8 waves; block owns 16 rows x 128 cols; each wave owns
// one 16x16 output tile and chains K/4 = 16 WMMAs. EXEC stays all-1s through
// the WMMA loop (loads are clamped, stores guarded afterwards).
// ---------------------------------------------------------------------------
__global__ __launch_bounds__(256) void feast_gemm_wmma_kernel(const float* __restrict__ X,
                                                              const float* __restrict__ W,
                                                              float* __restrict__ Y, int N) {
    const int wave = threadIdx.x >> 5;      // 0..7 : column tile
    const int lane = threadIdx.x & 31;
    const int l15  = lane & 15;
    const int hi   = lane >> 4;             // 0 or 1 (half-wave)
    const int row0 = blockIdx.x * 16;
    const int colB = wave * 16;

    // Clamp A row so out-of-range rows read row N-1 (stores are guarded).
    int arow = row0 + l15;
    if (arow >= N) arow = N - 1;
    const float* __restrict__ Arow = X + (size_t)arow * IN_CH;

    v8f acc = {};
#pragma unroll
    for (int k = 0; k < IN_CH; k += 4) {
        const int kk = k + hi * 2;          // this half-wave's K pair
        v2f a;
        a.x = Arow[kk];
        a.y = Arow[kk + 1];
        v2f b;
        b.x = W[kk * WCOLS + colB + l15];
        b.y = W[(kk + 1) * WCOLS + colB + l15];
        acc = __builtin_amdgcn_wmma_f32_16x16x4_f32(
            /*neg_a=*/false, a, /*neg_b=*/false, b,
            /*c_mod=*/(short)0, acc, /*reuse_a=*/false, /*reuse_b=*/false);
    }

#pragma unroll
    for (int r = 0; r < 8; ++r) {
        int row = row0 + r + hi * 8;        // D: VGPR r holds M=r (lanes 0-15), M=r+8 (16-31)
        if (row < N) Y[(size_t)row * WCOLS + colB + l15] = acc[r];
    }
}

// ---------------------------------------------------------------------------
// Edge phase: one wave32 per edge, lane = output channel.
//   q = softmax(P[src]-P[dst]+c); msg = sum_h q_h * Y[src, h*32 + lane]
//   atomicAdd agg[dst*32+lane], lane0 atomicAdds deg[dst].
// src==dst edges carry weight 0 in the reference -> skipped entirely.
// ---------------------------------------------------------------------------
__global__ __launch_bounds__(256) void feast_edge_kernel(const long long* __restrict__ ei,
                                                         const float4* __restrict__ P,
                                                         const float* __restrict__ Y,
                                                         const float* __restrict__ c,
                                                         float* __restrict__ agg,
                                                         float* __restrict__ deg,
                                                         long long E) {
    const int warp = threadIdx.x >> 5;
    const int lane = threadIdx.x & 31;
    long long e = (long long)blockIdx.x * 8 + warp;
    if (e >= E) return;
    long long src = ei[e];
    long long dst = ei[E + e];
    if (src == dst) return;                 // remove_self_loops -> weight 0

    float4 ps = P[src];
    float4 pd = P[dst];
    float4 cc = *(const float4*)c;
    float q0, q1, q2, q3;
    softmax4(ps.x - pd.x + cc.x, ps.y - pd.y + cc.y,
             ps.z - pd.z + cc.z, ps.w - pd.w + cc.w, q0, q1, q2, q3);

    const float* __restrict__ Yr = Y + (size_t)src * WCOLS;
    float msg = q0 * Yr[lane] + q1 * Yr[32 + lane] + q2 * Yr[64 + lane] + q3 * Yr[96 + lane];

    __hip_atomic_fetch_add(agg + (size_t)dst * OUT_CH + lane, msg,
                           __ATOMIC_RELAXED, __HIP_MEMORY_SCOPE_AGENT);
    if (lane == 0)
        __hip_atomic_fetch_add(deg + dst, 1.0f,
                               __ATOMIC_RELAXED, __HIP_MEMORY_SCOPE_AGENT);
}

// ---------------------------------------------------------------------------
// Finalize: one wave per node. Adds the self-loop message (q = softmax(c),
// a per-graph constant), divides by degree, adds bias, leaky-relu(0.2).
// ---------------------------------------------------------------------------
__global__ __launch_bounds__(256) void feast_final_kernel(const float* __restrict__ Y,
                                                          const float* __restrict__ agg,
                                                          const float* __restrict__ deg,
                                                          const float* __restrict__ c,
                                                          const float* __restrict__ b,
                                                          float* __restrict__ out, int N) {
    const int warp = threadIdx.x >> 5;
    const int lane = threadIdx.x & 31;
    int i = blockIdx.x * 8 + warp;
    if (i >= N) return;

    float4 cc = *(const float4*)c;
    float q0, q1, q2, q3;
    softmax4(cc.x, cc.y, cc.z, cc.w, q0, q1, q2, q3);

    const float* __restrict__ Yr = Y + (size_t)i * WCOLS;
    float selfMsg = q0 * Yr[lane] + q1 * Yr[32 + lane] + q2 * Yr[64 + lane] + q3 * Yr[96 + lane];

    float d = deg[i] + 1.0f;                // +1 for the added self loop
    float denom = fmaxf(d, 1.0f);
    float v = (agg[(size_t)i * OUT_CH + lane] + selfMsg) / denom + b[lane];
    out[(size_t)i * OUT_CH + lane] = (v > 0.0f) ? v : 0.2f * v;
}

// ---------------------------------------------------------------------------
static inline size_t align256(size_t x) { return (x + 255) & ~(size_t)255; }

extern "C" void kernel_launch(void* const* d_in, const int* in_sizes, int n_in,
                              void* d_out, int out_size, void* d_ws, size_t ws_size,
                              hipStream_t stream) {
    const float*     x[2]  = { (const float*)d_in[0],     (const float*)d_in[2]  };
    const long long* ei[2] = { (const long long*)d_in[1], (const long long*)d_in[3] };
    const float*     W[2]  = { (const float*)d_in[4],     (const float*)d_in[8]  };
    const float*     U[2]  = { (const float*)d_in[5],     (const float*)d_in[9]  };
    const float*     c[2]  = { (const float*)d_in[6],     (const float*)d_in[10] };
    const float*     b[2]  = { (const float*)d_in[7],     (const float*)d_in[11] };

    const int       N = in_sizes[0] / IN_CH;
    const long long E = in_sizes[1] / 2;

    // Workspace layout (reused across the two graphs, run sequentially):
    char* ws = (char*)d_ws;
    float*  Yw   = (float*)ws;                       ws += align256((size_t)N * WCOLS * sizeof(float));
    float4* Pw   = (float4*)ws;                      ws += align256((size_t)N * sizeof(float4));
    float*  aggw = (float*)ws;                       ws += align256((size_t)N * OUT_CH * sizeof(float));
    float*  degw = (float*)ws;

    const int pBlocks    = (N + 255) / 256;
    const int gemmBlocks = (N + 15) / 16;
    const int edgeBlocks = (int)((E + 7) / 8);
    const int nodeBlocks = (N + 7) / 8;

    for (int g = 0; g < 2; ++g) {
        hipMemsetAsync(aggw, 0, (size_t)N * OUT_CH * sizeof(float), stream);
        hipMemsetAsync(degw, 0, (size_t)N * sizeof(float), stream);

        feast_p_kernel<<<pBlocks, 256, 0, stream>>>(x[g], U[g], Pw, N);
        feast_gemm_wmma_kernel<<<gemmBlocks, 256, 0, stream>>>(x[g], W[g], Yw, N);
        feast_edge_kernel<<<edgeBlocks, 256, 0, stream>>>(ei[g], Pw, Yw, c[g], aggw, degw, E);
        feast_final_kernel<<<nodeBlocks, 256, 0, stream>>>(Yw, aggw, degw, c[g], b[g],
                                                           (float*)d_out + (size_t)g * N * OUT_CH, N);
    }
}